// GraphSAGE_59742995087911
// MI455X (gfx1250) — compile-verified
//
#include <hip/hip_runtime.h>
#include <hip/hip_bf16.h>

typedef float v2f __attribute__((ext_vector_type(2)));
typedef float v8f __attribute__((ext_vector_type(8)));

#define NNODES 100000
#define DIM 64

// ---------------------------------------------------------------------------
// Degree count: cnt[dst] += 1.0 (exact in fp32 for any realistic degree)
// ---------------------------------------------------------------------------
__global__ void count_edges_kernel(const int* __restrict__ dst,
                                   float* __restrict__ cnt, int n_edges) {
  int e = blockIdx.x * blockDim.x + threadIdx.x;
  if (e < n_edges) atomicAdd(&cnt[dst[e]], 1.0f);
}

// cnt[i] = 1 / max(cnt[i], 1)
__global__ void inv_count_kernel(float* __restrict__ cnt, int n) {
  int i = blockIdx.x * blockDim.x + threadIdx.x;
  if (i < n) cnt[i] = 1.0f / fmaxf(cnt[i], 1.0f);
}

// ---------------------------------------------------------------------------
// Edge scatter: agg[dst] += feat[src]. 16 threads per edge, float4 chunks.
// All traffic is L2-resident (feat/agg are 25.6MB each vs 192MB L2).
// ---------------------------------------------------------------------------
__global__ void scatter_feat_kernel(const float* __restrict__ feat,
                                    const int* __restrict__ src,
                                    const int* __restrict__ dst,
                                    float* __restrict__ agg, int n_edges) {
  int t = blockIdx.x * blockDim.x + threadIdx.x;
  int e = t >> 4;
  if (e >= n_edges) return;
  int c = (t & 15) << 2;
  int s = src[e], d = dst[e];
  const float4 v = *(const float4*)(feat + (size_t)s * DIM + c);
  float* o = agg + (size_t)d * DIM + c;
  atomicAdd(o + 0, v.x);
  atomicAdd(o + 1, v.y);
  atomicAdd(o + 2, v.z);
  atomicAdd(o + 3, v.w);
}

// ---------------------------------------------------------------------------
// SAGE layer: H = relu( (agg .* invc) @ Wl + X @ Wr + b ), via fp32 WMMA.
// One wave computes a 16-row x 64-col block (4 C tiles, 128 WMMA ops).
// FINAL=true fuses the classifier: out = relu(...) @ Wc + bc (no H store).
//
// Fragment layouts (CDNA5 ISA 7.12.2, wave32):
//   C/D 16x16 f32 : VGPR r, lane -> M = r + 8*(lane>>4), N = lane&15
//   A   16x4  f32 : lane -> M = lane&15; VGPR j -> K = 2*(lane>>4) + j
//   B    4x16 f32 : lane -> N = lane&15; VGPR j -> K = 2*(lane>>4) + j
// ---------------------------------------------------------------------------
template <bool FINAL>
__global__ __launch_bounds__(64) void sage_layer_kernel(
    const float* __restrict__ agg, const float* __restrict__ invc,
    const float* __restrict__ x,
    const float* __restrict__ Wl, const float* __restrict__ Wr,
    const float* __restrict__ bias,
    float* __restrict__ Hout,
    const float* __restrict__ Wc, const float* __restrict__ bc,
    float* __restrict__ out) {
  const int lane = threadIdx.x & 31;
  const int wave = threadIdx.x >> 5;
  const int rb   = blockIdx.x * 2 + wave;   // 16-row block index, 0..6249
  const int m0   = rb * 16;
  const int col  = lane & 15;               // N within tile / A-row within block
  const int half = lane >> 4;

  const int m = m0 + col;                   // A-matrix row handled by this lane
  const float ic = invc[m];
  const float* aggRow = agg + (size_t)m * DIM;
  const float* xRow   = x   + (size_t)m * DIM;

  v8f c0, c1, c2, c3;
  {
    const float b0 = bias[col], b1 = bias[16 + col];
    const float b2 = bias[32 + col], b3 = bias[48 + col];
#pragma unroll
    for (int r = 0; r < 8; ++r) { c0[r] = b0; c1[r] = b1; c2[r] = b2; c3[r] = b3; }
  }

#pragma unroll
  for (int k0 = 0; k0 < DIM; k0 += 4) {
    const int kb = k0 + 2 * half;           // this lane's K pair
    v2f Am, Ax, B;
    {
      const float2 a = *(const float2*)(aggRow + kb);
      Am[0] = a.x * ic; Am[1] = a.y * ic;   // mean = agg * invcnt, on the fly
      const float2 b = *(const float2*)(xRow + kb);
      Ax[0] = b.x; Ax[1] = b.y;
    }
    const float* wl = Wl + (size_t)kb * DIM + col;
    const float* wr = Wr + (size_t)kb * DIM + col;

    // tile 0 (N 0..15)
    B[0] = wl[0];       B[1] = wl[DIM];
    c0 = __builtin_amdgcn_wmma_f32_16x16x4_f32(false, Am, false, B, (short)0, c0, false, false);
    B[0] = wr[0];       B[1] = wr[DIM];
    c0 = __builtin_amdgcn_wmma_f32_16x16x4_f32(false, Ax, false, B, (short)0, c0, false, false);
    // tile 1 (N 16..31)
    B[0] = wl[16];      B[1] = wl[DIM + 16];
    c1 = __builtin_amdgcn_wmma_f32_16x16x4_f32(false, Am, false, B, (short)0, c1, false, false);
    B[0] = wr[16];      B[1] = wr[DIM + 16];
    c1 = __builtin_amdgcn_wmma_f32_16x16x4_f32(false, Ax, false, B, (short)0, c1, false, false);
    // tile 2 (N 32..47)
    B[0] = wl[32];      B[1] = wl[DIM + 32];
    c2 = __builtin_amdgcn_wmma_f32_16x16x4_f32(false, Am, false, B, (short)0, c2, false, false);
    B[0] = wr[32];      B[1] = wr[DIM + 32];
    c2 = __builtin_amdgcn_wmma_f32_16x16x4_f32(false, Ax, false, B, (short)0, c2, false, false);
    // tile 3 (N 48..63)
    B[0] = wl[48];      B[1] = wl[DIM + 48];
    c3 = __builtin_amdgcn_wmma_f32_16x16x4_f32(false, Am, false, B, (short)0, c3, false, false);
    B[0] = wr[48];      B[1] = wr[DIM + 48];
    c3 = __builtin_amdgcn_wmma_f32_16x16x4_f32(false, Ax, false, B, (short)0, c3, false, false);
  }

  // ReLU
#pragma unroll
  for (int r = 0; r < 8; ++r) {
    c0[r] = fmaxf(c0[r], 0.0f);
    c1[r] = fmaxf(c1[r], 0.0f);
    c2[r] = fmaxf(c2[r], 0.0f);
    c3[r] = fmaxf(c3[r], 0.0f);
  }

  if (!FINAL) {
#pragma unroll
    for (int r = 0; r < 8; ++r) {
      const size_t row = (size_t)(m0 + r + 8 * half) * DIM;
      Hout[row + col]      = c0[r];
      Hout[row + 16 + col] = c1[r];
      Hout[row + 32 + col] = c2[r];
      Hout[row + 48 + col] = c3[r];
    }
  } else {
    // Fused classifier: per-lane partial dot with Wc, butterfly-reduce the
    // 16-lane half, lane {0,16} writes 8 logits each.
    const float w0 = Wc[col], w1 = Wc[16 + col], w2 = Wc[32 + col], w3 = Wc[48 + col];
    float p[8];
#pragma unroll
    for (int r = 0; r < 8; ++r)
      p[r] = c0[r] * w0 + c1[r] * w1 + c2[r] * w2 + c3[r] * w3;
#pragma unroll
    for (int r = 0; r < 8; ++r) {
      p[r] += __shfl_xor(p[r], 1, 32);
      p[r] += __shfl_xor(p[r], 2, 32);
      p[r] += __shfl_xor(p[r], 4, 32);
      p[r] += __shfl_xor(p[r], 8, 32);
    }
    if (col == 0) {
      const float bcv = bc[0];
#pragma unroll
      for (int r = 0; r < 8; ++r)
        out[m0 + r + 8 * half] = p[r] + bcv;
    }
  }
}

// ---------------------------------------------------------------------------
extern "C" void kernel_launch(void* const* d_in, const int* in_sizes, int n_in,
                              void* d_out, int out_size, void* d_ws, size_t ws_size,
                              hipStream_t stream) {
  const float* x   = (const float*)d_in[0];
  const int*   ei  = (const int*)d_in[1];
  const float* W1l = (const float*)d_in[2];
  const float* W1r = (const float*)d_in[3];
  const float* b1  = (const float*)d_in[4];
  const float* W2l = (const float*)d_in[5];
  const float* W2r = (const float*)d_in[6];
  const float* b2  = (const float*)d_in[7];
  const float* Wc  = (const float*)d_in[8];
  const float* bc  = (const float*)d_in[9];
  float* out = (float*)d_out;

  const int n_edges = in_sizes[1] / 2;
  const int* src = ei;
  const int* dst = ei + n_edges;

  const size_t featBytes = (size_t)NNODES * DIM * sizeof(float);
  char* ws = (char*)d_ws;
  float* agg = (float*)ws;                      // 25.6 MB, reused both layers
  float* h1  = (float*)(ws + featBytes);        // 25.6 MB
  float* cnt = (float*)(ws + 2 * featBytes);    // 0.4 MB

  // ---- degree counts (shared by both layers) ----
  hipMemsetAsync(cnt, 0, (size_t)NNODES * sizeof(float), stream);
  hipMemsetAsync(agg, 0, featBytes, stream);
  count_edges_kernel<<<(n_edges + 255) / 256, 256, 0, stream>>>(dst, cnt, n_edges);
  inv_count_kernel<<<(NNODES + 255) / 256, 256, 0, stream>>>(cnt, NNODES);

  // ---- layer 1 ----
  const int scatterBlocks = (n_edges * 16 + 255) / 256;
  scatter_feat_kernel<<<scatterBlocks, 256, 0, stream>>>(x, src, dst, agg, n_edges);
  sage_layer_kernel<false><<<NNODES / 32, 64, 0, stream>>>(
      agg, cnt, x, W1l, W1r, b1, h1, nullptr, nullptr, nullptr);

  // ---- layer 2 + fused classifier ----
  hipMemsetAsync(agg, 0, featBytes, stream);
  scatter_feat_kernel<<<scatterBlocks, 256, 0, stream>>>(h1, src, dst, agg, n_edges);
  sage_layer_kernel<true><<<NNODES / 32, 64, 0, stream>>>(
      agg, cnt, h1, W2l, W2r, b2, nullptr, Wc, bc, out);
}